// AtomicBasis_50319836840565
// MI455X (gfx1250) — compile-verified
//
#include <hip/hip_runtime.h>
#include <math.h>

typedef __attribute__((ext_vector_type(2))) float v2f;
typedef __attribute__((ext_vector_type(8))) float v8f;

#define THREADS   128
#define PER_BLOCK 2048   // neighbors per block, multiple of 64
#define N_OUT     1728   // 64 k * 27 (x,y,z)

__device__ __forceinline__ v8f wmma_f32_16x16x4(v2f a, v2f b, v8f c) {
  // D = A(16x4) * B(4x16) + C(16x16), all f32
  return __builtin_amdgcn_wmma_f32_16x16x4_f32(
      /*neg_a=*/false, a, /*neg_b=*/false, b,
      /*c_mod=*/(short)0, c, /*reuse_a=*/false, /*reuse_b=*/false);
}

__global__ __launch_bounds__(THREADS) void ab_partial_kernel(
    const float* __restrict__ h,     // (nA, 64)
    const float* __restrict__ rel,   // (3, nA)
    const float* __restrict__ W,     // (64, 64)
    float* __restrict__ partials,    // (gridDim.x, 1728)
    int nA, int perBlock)
{
  // Padded LDS (stride 68 floats = 272B keeps 16B alignment, breaks bank conflicts)
  __shared__ float sW  [64 * 68];  // W[k][j]
  __shared__ float sRad[64 * 68];  // radial[n][a_local]
  __shared__ float sWt [64 * 68];  // w[k][a_local] (radial * hW, transposed)
  __shared__ float sT  [64 * 40];  // T[a_local][m], m in [0,32), cols 27..31 zero

  const int tid  = threadIdx.x;
  const int lane = tid & 31;
  const int wv   = tid >> 5;     // wave id 0..3
  const int l16  = lane & 15;
  const int hi   = lane >> 4;    // 0 for lanes 0-15, 1 for lanes 16-31

  // ---- stage 0: W (64x64) -> LDS, coalesced float4 ----
  for (int it = 0; it < 8; ++it) {
    int idx = (it * THREADS + tid) * 4;        // 0..4095
    int row = idx >> 6, col = idx & 63;
    float4 w4 = *(const float4*)(W + row * 64 + col);
    *(float4*)(sW + row * 68 + col) = w4;
  }
  __syncthreads();

  const int aStart = blockIdx.x * perBlock;
  v8f accD[2] = {v8f{}, v8f{}};                // 2 (kt,mt) output tiles per wave

  for (int t0 = 0; t0 < perBlock; t0 += 64) {
    const int aBase = aStart + t0;

    // ---- per-neighbor geometry + radial basis (1 thread / neighbor) ----
    if (tid < 64) {
      int a   = aBase + tid;
      bool ok = (a < nA);
      int ac  = ok ? a : (nA - 1);
      float x = rel[ac], y = rel[nA + ac], z = rel[2 * nA + ac];
      float r    = sqrtf(x * x + y * y + z * z);
      float invr = 1.0f / r;
      float n3[3] = {x * invr, y * invr, z * invr};
      #pragma unroll
      for (int m = 0; m < 27; ++m) {
        float v = n3[m / 9] * n3[(m / 3) % 3] * n3[m % 3];
        sT[tid * 40 + m] = ok ? v : 0.0f;
      }
      #pragma unroll
      for (int m = 27; m < 32; ++m) sT[tid * 40 + m] = 0.0f;

      // radial[n] = sqrt(2/5) * sin((n+1)*pi*r/5) / r  via Chebyshev recurrence
      float theta = 0.6283185307179586f * r;   // pi/5 * r
      float s, c;
      sincosf(theta, &s, &c);
      float coef = ok ? (0.6324555320336759f * invr) : 0.0f;  // sqrt(0.4)/r
      float c2 = 2.0f * c, sm1 = 0.0f, sn = s;
      #pragma unroll
      for (int n = 0; n < 64; ++n) {
        sRad[n * 68 + tid] = coef * sn;
        float nxt = c2 * sn - sm1;
        sm1 = sn; sn = nxt;
      }
    }
    __syncthreads();

    // ---- stage A: hW tile (16 neighbors x 64 ch) per wave via f32 WMMA ----
    // A-frag read directly from global h (each 256B row fully consumed).
    {
      const int aw = wv * 16;                  // wave's local neighbor base
      int arow = aBase + aw + l16;
      if (arow >= nA) arow = nA - 1;
      const float* hrow = h + (size_t)arow * 64;
      #pragma unroll
      for (int kt = 0; kt < 4; ++kt) {
        v8f acc = {};
        #pragma unroll
        for (int kk = 0; kk < 16; ++kk) {
          int j = 4 * kk + 2 * hi;             // K-pair this half-wave owns
          v2f af = *(const v2f*)(hrow + j);                      // h[a][j], h[a][j+1]
          v2f bf = *(const v2f*)(sW + (16 * kt + l16) * 68 + j); // W[k][j], W[k][j+1]
          acc = wmma_f32_16x16x4(af, bf, acc);
        }
        // D layout: M(=a_local) = v + 8*hi, N(=k in tile) = l16
        #pragma unroll
        for (int v = 0; v < 8; ++v) {
          int aL = aw + v + 8 * hi;
          int k  = 16 * kt + l16;
          sWt[k * 68 + aL] = acc[v] * sRad[k * 68 + aL];
        }
      }
    }
    __syncthreads();

    // ---- stage D: A[k,m] += w(64k x 64a) * T(64a x 32m); 2 tiles per wave ----
    #pragma unroll
    for (int i = 0; i < 2; ++i) {
      const int p  = wv * 2 + i;
      const int kt = p >> 1, mt = p & 1;
      v8f acc = accD[i];
      #pragma unroll
      for (int aa = 0; aa < 16; ++aa) {
        int a0 = 4 * aa + 2 * hi;              // K-pair this half-wave owns
        v2f af = *(const v2f*)(sWt + (16 * kt + l16) * 68 + a0); // w[k][a0], w[k][a0+1]
        v2f bf;
        bf.x = sT[a0 * 40 + 16 * mt + l16];                      // T[a0][m]
        bf.y = sT[(a0 + 1) * 40 + 16 * mt + l16];                // T[a0+1][m]
        acc = wmma_f32_16x16x4(af, bf, acc);
      }
      accD[i] = acc;
    }
    __syncthreads();   // sT/sRad/sWt reused next iteration
  }

  // ---- write per-block partials (each (k,m) owned by exactly one lane/vgpr) ----
  float* pb = partials + (size_t)blockIdx.x * N_OUT;
  #pragma unroll
  for (int i = 0; i < 2; ++i) {
    const int p  = wv * 2 + i;
    const int kt = p >> 1, mt = p & 1;
    #pragma unroll
    for (int v = 0; v < 8; ++v) {
      int k = 16 * kt + v + 8 * hi;
      int m = 16 * mt + l16;
      if (m < 27) pb[k * 27 + m] = accD[i][v];
    }
  }
}

__global__ void ab_reduce_kernel(const float* __restrict__ partials,
                                 float* __restrict__ out, int nBlocks)
{
  int e = blockIdx.x * blockDim.x + threadIdx.x;
  if (e >= N_OUT) return;
  float s = 0.0f;
  for (int b = 0; b < nBlocks; ++b) s += partials[(size_t)b * N_OUT + e];
  out[e] = s;
}

extern "C" void kernel_launch(void* const* d_in, const int* in_sizes, int n_in,
                              void* d_out, int out_size, void* d_ws, size_t ws_size,
                              hipStream_t stream) {
  const float* h   = (const float*)d_in[0];   // (nA, 64)
  const float* rel = (const float*)d_in[1];   // (3, nA)
  const float* W   = (const float*)d_in[2];   // (64, 64)
  float* out = (float*)d_out;

  int nA = in_sizes[0] / 64;
  if (nA < 1) nA = 1;

  int perBlock = PER_BLOCK;
  int nBlocks  = (nA + perBlock - 1) / perBlock;
  // fit partials into workspace if it is small
  size_t need = (size_t)nBlocks * N_OUT * sizeof(float);
  if (need > ws_size) {
    int maxB = (int)(ws_size / (N_OUT * sizeof(float)));
    if (maxB < 1) maxB = 1;
    nBlocks  = maxB;
    perBlock = (((nA + nBlocks - 1) / nBlocks) + 63) & ~63;
    nBlocks  = (nA + perBlock - 1) / perBlock;
  }

  float* partials = (float*)d_ws;
  ab_partial_kernel<<<nBlocks, THREADS, 0, stream>>>(h, rel, W, partials, nA, perBlock);
  ab_reduce_kernel<<<(N_OUT + 255) / 256, 256, 0, stream>>>(partials, out, nBlocks);
}